// SuperCLAPTrainer_24936580121220
// MI455X (gfx1250) — compile-verified
//
#include <hip/hip_runtime.h>
#include <math.h>

typedef __attribute__((ext_vector_type(2))) float v2f;
typedef __attribute__((ext_vector_type(8))) float v8f;

#define Bn    16
#define T_TOK 1024
#define T_PH  2048
#define T_AUD 1500
#define Dd    512
#define Wn    256

#define NCHUNK 32          // split-K chunks per batch for the audio reduction
#define CHUNK_ROWS 47      // ceil(1500/32)

// workspace layout (in floats)
#define WS_AUDPART 0                       // 16 * 32 * 512 partial sums (1 MB)
#define WS_METAIN  (Bn * NCHUNK * Dd)      // 16 * 512 meta_input rows
#define WS_METARAW (WS_METAIN + Bn * Dd)   // 16 * 512 pre-norm meta embedding

// ---------------------------------------------------------------------------
// Kernel 1a: split-K partial sums over audio_outputs rows (the 49MB read).
// grid = 512 blocks (16 batches x 32 chunks), 128 threads (float4 each).
// 4 independent accumulators -> 4 outstanding b128 loads per thread.
// ---------------------------------------------------------------------------
__global__ void k_audio_partial(const float* __restrict__ audio,
                                float* __restrict__ ws) {
  const int b     = blockIdx.x >> 5;
  const int chunk = blockIdx.x & 31;
  const int t     = threadIdx.x;             // 0..127 -> cols 4t..4t+3
  const int r0 = chunk * CHUNK_ROWS;
  const int r1 = (r0 + CHUNK_ROWS < T_AUD) ? (r0 + CHUNK_ROWS) : T_AUD;
  const float4* __restrict__ src =
      (const float4*)audio + (size_t)b * T_AUD * (Dd / 4) + t;

  float4 a0 = make_float4(0.f, 0.f, 0.f, 0.f);
  float4 a1 = a0, a2 = a0, a3 = a0;
  int r = r0;
  for (; r + 4 <= r1; r += 4) {
    float4 v0 = src[(size_t)(r + 0) * (Dd / 4)];
    float4 v1 = src[(size_t)(r + 1) * (Dd / 4)];
    float4 v2 = src[(size_t)(r + 2) * (Dd / 4)];
    float4 v3 = src[(size_t)(r + 3) * (Dd / 4)];
    a0.x += v0.x; a0.y += v0.y; a0.z += v0.z; a0.w += v0.w;
    a1.x += v1.x; a1.y += v1.y; a1.z += v1.z; a1.w += v1.w;
    a2.x += v2.x; a2.y += v2.y; a2.z += v2.z; a2.w += v2.w;
    a3.x += v3.x; a3.y += v3.y; a3.z += v3.z; a3.w += v3.w;
  }
  for (; r < r1; ++r) {
    float4 v = src[(size_t)r * (Dd / 4)];
    a0.x += v.x; a0.y += v.y; a0.z += v.z; a0.w += v.w;
  }
  float4 acc;
  acc.x = (a0.x + a1.x) + (a2.x + a3.x);
  acc.y = (a0.y + a1.y) + (a2.y + a3.y);
  acc.z = (a0.z + a1.z) + (a2.z + a3.z);
  acc.w = (a0.w + a1.w) + (a2.w + a3.w);
  ((float4*)(ws + WS_AUDPART))[(b * NCHUNK + chunk) * (Dd / 4) + t] = acc;
}

// ---------------------------------------------------------------------------
// Kernel 1b: reduce 32 partials, mean, L2-normalize -> audio_emb (d_out+8192).
// grid = 16 blocks, 128 threads.
// ---------------------------------------------------------------------------
__global__ void k_audio_finish(const float* __restrict__ ws,
                               float* __restrict__ out) {
  const int b = blockIdx.x;
  const int t = threadIdx.x;
  const float4* part =
      (const float4*)(ws + WS_AUDPART) + (size_t)b * NCHUNK * (Dd / 4) + t;
  float4 s = make_float4(0.f, 0.f, 0.f, 0.f);
#pragma unroll
  for (int c = 0; c < NCHUNK; ++c) {
    float4 v = part[c * (Dd / 4)];
    s.x += v.x; s.y += v.y; s.z += v.z; s.w += v.w;
  }
  const float invT = 1.0f / (float)T_AUD;
  s.x *= invT; s.y *= invT; s.z *= invT; s.w *= invT;

  __shared__ float red[128];
  red[t] = s.x * s.x + s.y * s.y + s.z * s.z + s.w * s.w;
  __syncthreads();
  for (int off = 64; off > 0; off >>= 1) {
    if (t < off) red[t] += red[t + off];
    __syncthreads();
  }
  const float inv = 1.0f / sqrtf(red[0]);
  float4 o = make_float4(s.x * inv, s.y * inv, s.z * inv, s.w * inv);
  ((float4*)(out + Bn * Dd))[b * (Dd / 4) + t] = o;
}

// ---------------------------------------------------------------------------
// Kernel 2: build the ONE meta_input row per batch that is actually used:
// phoneme_outputs[b, p] + (valid && tok_len>0) * word_mean(word containing p).
// grid = 16 blocks, 128 threads.
// ---------------------------------------------------------------------------
__global__ void k_meta_input(const float* __restrict__ phoneme,
                             const float* __restrict__ token,
                             const int* __restrict__ wtl,
                             const int* __restrict__ wpl,
                             const int* __restrict__ pidx,
                             float* __restrict__ ws) {
  const int b = blockIdx.x;
  const int t = threadIdx.x;
  __shared__ int sh_p, sh_start, sh_len;
  if (t == 0) {
    const int p = pidx[b];
    const int* phLen = wpl + b * Wn;
    const int* tkLen = wtl + b * Wn;
    int cum = 0, w = 0;
    for (int i = 0; i < Wn; ++i) {           // searchsorted(cum, p, 'right')
      cum += phLen[i];
      if (cum <= p) ++w;
    }
    const int valid = (p < cum);
    const int wc = (w < Wn - 1) ? w : (Wn - 1);
    int tstart = 0;
    for (int i = 0; i < wc; ++i) tstart += tkLen[i];
    const int tlen = tkLen[wc];
    sh_p = p;
    sh_start = tstart;
    sh_len = (valid && tlen > 0) ? tlen : 0;
  }
  __syncthreads();

  const float4* ph4 = (const float4*)phoneme;
  const float4* tk4 = (const float4*)token;
  float4 v = ph4[((size_t)b * T_PH + sh_p) * (Dd / 4) + t];
  const int len = sh_len;
  if (len > 0) {
    float4 s = make_float4(0.f, 0.f, 0.f, 0.f);
    for (int r = 0; r < len; ++r) {
      float4 tv = tk4[((size_t)b * T_TOK + sh_start + r) * (Dd / 4) + t];
      s.x += tv.x; s.y += tv.y; s.z += tv.z; s.w += tv.w;
    }
    const float invc = 1.0f / (float)len;
    v.x += s.x * invc; v.y += s.y * invc; v.z += s.z * invc; v.w += s.w * invc;
  }
  ((float4*)(ws + WS_METAIN))[b * (Dd / 4) + t] = v;
}

// ---------------------------------------------------------------------------
// Kernel 3: WMMA GEMM  metaRaw[16,512] = meta_input[16,512] @ W_meta + b_meta
// V_WMMA_F32_16X16X4_F32, 1 block x 8 waves, each wave owns 64 output cols
// (4 tiles of 16) with 4 independent accumulator chains for ILP.
// A-frag layout (ISA 7.12.2, 32b A 16x4): lane<16 -> {K0,K1}, lane>=16 -> {K2,K3}.
// B-frag layout (32b B 4x16): VGPR0 = rows K0/K2, VGPR1 = rows K1/K3.
// ---------------------------------------------------------------------------
__global__ void k_wmma_gemm(const float* __restrict__ ws_in,
                            const float* __restrict__ Wm,
                            const float* __restrict__ bm,
                            float* __restrict__ ws_out) {
  __shared__ float lA[Bn * Dd];                    // 32 KB: A staged in LDS
  {
    float4* l4 = (float4*)lA;
    const float4* m4 = (const float4*)(ws_in + WS_METAIN);
    for (int i = threadIdx.x; i < (Bn * Dd) / 4; i += 256) l4[i] = m4[i];
  }
  __syncthreads();

  const int lane = threadIdx.x & 31;
  const int wave = threadIdx.x >> 5;
  const int mRow = lane & 15;
  const int kSel = (lane >> 4) << 1;               // 0 or 2
  const int nBase = wave * 64 + (lane & 15);       // col for this lane, tile 0

  v8f acc0 = {0.f, 0.f, 0.f, 0.f, 0.f, 0.f, 0.f, 0.f};
  v8f acc1 = acc0, acc2 = acc0, acc3 = acc0;

  for (int k0 = 0; k0 < Dd; k0 += 4) {
    const int kb = k0 + kSel;
    v2f a = *(const v2f*)(&lA[mRow * Dd + kb]);    // ds_read_b64, 8B aligned
    const float* w0 = Wm + (size_t)kb * Dd + nBase;
    const float* w1 = w0 + Dd;
    v2f b0, b1, b2, b3;
    b0.x = w0[0];  b0.y = w1[0];
    b1.x = w0[16]; b1.y = w1[16];
    b2.x = w0[32]; b2.y = w1[32];
    b3.x = w0[48]; b3.y = w1[48];
    acc0 = __builtin_amdgcn_wmma_f32_16x16x4_f32(false, a, false, b0, (short)0, acc0, false, false);
    acc1 = __builtin_amdgcn_wmma_f32_16x16x4_f32(false, a, false, b1, (short)0, acc1, false, false);
    acc2 = __builtin_amdgcn_wmma_f32_16x16x4_f32(false, a, false, b2, (short)0, acc2, false, false);
    acc3 = __builtin_amdgcn_wmma_f32_16x16x4_f32(false, a, false, b3, (short)0, acc3, false, false);
  }

  // C/D layout: VGPR r -> rows r (lanes 0-15) and r+8 (lanes 16-31)
  float* metaRaw = ws_out + WS_METARAW;
  const int mHi = (lane >> 4) * 8;
  const float bias0 = bm[nBase + 0];
  const float bias1 = bm[nBase + 16];
  const float bias2 = bm[nBase + 32];
  const float bias3 = bm[nBase + 48];
#pragma unroll
  for (int r = 0; r < 8; ++r) {
    const int M = r + mHi;
    metaRaw[M * Dd + nBase + 0]  = acc0[r] + bias0;
    metaRaw[M * Dd + nBase + 16] = acc1[r] + bias1;
    metaRaw[M * Dd + nBase + 32] = acc2[r] + bias2;
    metaRaw[M * Dd + nBase + 48] = acc3[r] + bias3;
  }
}

// ---------------------------------------------------------------------------
// Kernel 4: normalize meta rows, 16x16 logits, dual log-softmax diag, loss.
// 1 block x 256 threads; all reductions fixed-order (deterministic).
// ---------------------------------------------------------------------------
__global__ void k_final(const float* __restrict__ ws,
                        const float* __restrict__ lscale,
                        float* __restrict__ out) {
  const int tid = threadIdx.x;
  const float* metaRaw = ws + WS_METARAW;
  const float* aud = out + Bn * Dd;                // audio_emb written earlier
  __shared__ float sInv[16];
  __shared__ float sLog[16][16];
  __shared__ float sLs[32];

  if (tid < 16) {
    float ss = 0.f;
    for (int k = 0; k < Dd; ++k) {
      const float v = metaRaw[tid * Dd + k];
      ss += v * v;
    }
    sInv[tid] = 1.0f / sqrtf(ss);
  }
  __syncthreads();

  // write normalized meta_emb -> d_out[0 .. 8191]
  for (int i = tid; i < Bn * Dd; i += 256) out[i] = metaRaw[i] * sInv[i >> 9];

  // logits[i][j] = scale * <audio_emb[i], meta_emb[j]>
  {
    const int i = tid >> 4, j = tid & 15;
    float dot = 0.f;
    for (int k = 0; k < Dd; ++k) dot += aud[i * Dd + k] * metaRaw[j * Dd + k];
    const float scale = expf(lscale[0]);
    sLog[i][j] = scale * dot * sInv[j];
  }
  __syncthreads();

  if (tid < 16) {                                  // rows of logits
    float m = -1e30f;
    for (int j = 0; j < 16; ++j) m = fmaxf(m, sLog[tid][j]);
    float s = 0.f;
    for (int j = 0; j < 16; ++j) s += expf(sLog[tid][j] - m);
    sLs[tid] = sLog[tid][tid] - m - logf(s);
  } else if (tid < 32) {                           // columns (logits.T rows)
    const int j = tid - 16;
    float m = -1e30f;
    for (int i = 0; i < 16; ++i) m = fmaxf(m, sLog[i][j]);
    float s = 0.f;
    for (int i = 0; i < 16; ++i) s += expf(sLog[i][j] - m);
    sLs[tid] = sLog[j][j] - m - logf(s);
  }
  __syncthreads();

  if (tid == 0) {
    float a = 0.f, mm = 0.f;
    for (int i = 0; i < 16; ++i) { a += sLs[i]; mm += sLs[16 + i]; }
    out[2 * Bn * Dd] = -(a * (1.0f / 16.f) + mm * (1.0f / 16.f)) * 0.5f;
  }
}

// ---------------------------------------------------------------------------
extern "C" void kernel_launch(void* const* d_in, const int* in_sizes, int n_in,
                              void* d_out, int out_size, void* d_ws,
                              size_t ws_size, hipStream_t stream) {
  (void)in_sizes; (void)n_in; (void)out_size; (void)ws_size;
  const float* tok  = (const float*)d_in[0];   // token_outputs   [16,1024,512]
  const float* ph   = (const float*)d_in[1];   // phoneme_outputs [16,2048,512]
  const float* aud  = (const float*)d_in[2];   // audio_outputs   [16,1500,512]
  const float* Wm   = (const float*)d_in[3];   // W_meta          [512,512]
  const float* bm   = (const float*)d_in[4];   // b_meta          [512]
  const float* ls   = (const float*)d_in[5];   // logit_scale     [1]
  const int*   wtl  = (const int*)d_in[6];     // word_token_lengths   [16,256]
  const int*   wpl  = (const int*)d_in[7];     // word_phoneme_lengths [16,256]
  const int*   pidx = (const int*)d_in[8];     // phonemes_index  [16]
  float* out = (float*)d_out;                  // [meta 8192 | audio 8192 | loss]
  float* ws  = (float*)d_ws;

  k_audio_partial<<<Bn * NCHUNK, 128, 0, stream>>>(aud, ws);
  k_audio_finish <<<Bn,          128, 0, stream>>>(ws, out);
  k_meta_input   <<<Bn,          128, 0, stream>>>(ph, tok, wtl, wpl, pidx, ws);
  k_wmma_gemm    <<<1,           256, 0, stream>>>(ws, Wm, bm, ws);
  k_final        <<<1,           256, 0, stream>>>(ws, ls, out);
}